// GATStack_33732673143362
// MI455X (gfx1250) — compile-verified
//
#include <hip/hip_runtime.h>
#include <hip/hip_bf16.h>

// ---------------- constants (match reference) ----------------
#define N_NODES 100000
#define N_EDGES 1600000
#define HID 64
#define N_LAYERS 16
#define N_GRAPHS 64
#define NEG_SLOPE 0.2f
#define BN_EPS 1e-5f

typedef float v2f __attribute__((ext_vector_type(2)));
typedef float v8f __attribute__((ext_vector_type(8)));
typedef int   v4i __attribute__((ext_vector_type(4)));
typedef __attribute__((address_space(1))) v4i gv4i;   // global int4
typedef __attribute__((address_space(3))) v4i lv4i;   // LDS int4

// Probe the CDNA5 async global->LDS path; fall back to plain LDS stores if the
// toolchain doesn't expose the builtins (compile-safety first).
#if defined(__has_builtin)
#  if __has_builtin(__builtin_amdgcn_global_load_async_to_lds_b128) && \
      __has_builtin(__builtin_amdgcn_s_wait_asynccnt)
#    define USE_ASYNC_LDS 1
#  endif
#endif
#ifndef USE_ASYNC_LDS
#  define USE_ASYNC_LDS 0
#endif

// ---------------- helpers ----------------
__device__ __forceinline__ void atomicMaxFloat(float* addr, float v) {
    // classic bit-pattern trick: int-max for non-negative, uint-min for negative
    if (v >= 0.0f) atomicMax((int*)addr, __float_as_int(v));
    else           atomicMin((unsigned int*)addr, __float_as_uint(v));
}

__device__ __forceinline__ float lrelu(float v) {
    return v >= 0.0f ? v : NEG_SLOPE * v;
}

// ---------------- GEMM: hp[n,64] = h[n,64] @ W[64,64] via V_WMMA_F32_16X16X4_F32 ----------------
// One wave per 16-row tile; wave computes all 64 output columns (4 x 16x16 tiles).
// W (16KB) is staged once per block into LDS (async global->LDS when available),
// so the B operands come from ds_load instead of 8x-replicated global loads.
// 32-bit A 16x4 layout: lanes 0-15 -> K=0(v0),K=1(v1); lanes 16-31 -> K=2(v0),K=3(v1).
// 32-bit B 4x16 layout: lane n (0-15) -> K=0(v0),K=1(v1); lane n+16 -> K=2(v0),K=3(v1).
__global__ __launch_bounds__(256) void gemm64_wmma(const float* __restrict__ h,
                                                   const float* __restrict__ W,
                                                   float* __restrict__ hp, int n) {
    __shared__ float Wl[HID * HID];          // 16 KB of the WGP's 320 KB LDS
    const int t = threadIdx.x;

#if USE_ASYNC_LDS
    {
        // each of the 256 threads copies 16 floats (4 x B128 async transfers)
        const float* gsrc = W + t * 16;
        float* ldst = &Wl[t * 16];
        #pragma unroll
        for (int i = 0; i < 4; ++i) {
            __builtin_amdgcn_global_load_async_to_lds_b128(
                (gv4i*)(gsrc + i * 4),
                (lv4i*)(ldst + i * 4),
                0, 0);
        }
        __builtin_amdgcn_s_wait_asynccnt(0);
    }
#else
    #pragma unroll
    for (int i = 0; i < 16; ++i) Wl[t + i * 256] = W[t + i * 256];
#endif
    __syncthreads();

    const int wave = blockIdx.x * (blockDim.x >> 5) + (t >> 5);
    const int lane = t & 31;
    const int rowBase = wave * 16;
    if (rowBase >= n) return;                // wave-uniform: EXEC all-ones for WMMA
    const int half = lane >> 4;              // 0: lanes 0-15, 1: lanes 16-31
    const int l16  = lane & 15;

    v8f c0 = {}, c1 = {}, c2 = {}, c3 = {};
    const float* arow = h + (size_t)(rowBase + l16) * HID;

    #pragma unroll
    for (int kk = 0; kk < 16; ++kk) {
        const int k0 = kk * 4 + half * 2;    // this lane's two K indices: k0, k0+1
        v2f a;
        a.x = arow[k0];
        a.y = arow[k0 + 1];
        const float* w0 = &Wl[k0 * HID];     // row k0 of W (LDS)
        const float* w1 = w0 + HID;          // row k0+1 of W (LDS)
        v2f b;
        b.x = w0[l16 +  0]; b.y = w1[l16 +  0];
        c0 = __builtin_amdgcn_wmma_f32_16x16x4_f32(false, a, false, b, (short)0, c0, false, false);
        b.x = w0[l16 + 16]; b.y = w1[l16 + 16];
        c1 = __builtin_amdgcn_wmma_f32_16x16x4_f32(false, a, false, b, (short)0, c1, false, false);
        b.x = w0[l16 + 32]; b.y = w1[l16 + 32];
        c2 = __builtin_amdgcn_wmma_f32_16x16x4_f32(false, a, false, b, (short)0, c2, false, false);
        b.x = w0[l16 + 48]; b.y = w1[l16 + 48];
        c3 = __builtin_amdgcn_wmma_f32_16x16x4_f32(false, a, false, b, (short)0, c3, false, false);
    }

    // C/D layout: VGPR r, lanes 0-15 -> M=r; lanes 16-31 -> M=r+8; N = lane&15 (+16*ct)
    #pragma unroll
    for (int r = 0; r < 8; ++r) {
        const size_t row = (size_t)(rowBase + r + 8 * half);
        float* o = hp + row * HID + l16;
        o[ 0] = c0[r];
        o[16] = c1[r];
        o[32] = c2[r];
        o[48] = c3[r];
    }
}

// ---------------- per-row attention dots: e_src = hp.a_s, e_dst = hp.a_d ----------------
__global__ __launch_bounds__(256) void dots_kernel(const float* __restrict__ hp,
                                                   const float* __restrict__ a_s,
                                                   const float* __restrict__ a_d,
                                                   float* __restrict__ es,
                                                   float* __restrict__ ed, int n) {
    const int warp = (blockIdx.x * blockDim.x + threadIdx.x) >> 5;
    const int lane = threadIdx.x & 31;
    if (warp >= n) return;
    const float* row = hp + (size_t)warp * HID;
    float x0 = row[lane], x1 = row[lane + 32];
    float s1 = x0 * a_s[lane] + x1 * a_s[lane + 32];
    float s2 = x0 * a_d[lane] + x1 * a_d[lane + 32];
    #pragma unroll
    for (int off = 16; off > 0; off >>= 1) {
        s1 += __shfl_xor(s1, off, 32);
        s2 += __shfl_xor(s2, off, 32);
    }
    if (lane == 0) { es[warp] = s1; ed[warp] = s2; }
}

// ---------------- per-layer init: m=-inf, z=0, out=bias broadcast, stats=0 ----------------
__global__ __launch_bounds__(256) void init_layer(float* __restrict__ m, float* __restrict__ z,
                                                  float* __restrict__ out,
                                                  const float* __restrict__ bias,
                                                  float* __restrict__ stats, int n) {
    const int idx = blockIdx.x * blockDim.x + threadIdx.x;
    if (idx < 128) stats[idx] = 0.0f;
    if (idx >= n * HID) return;
    const int i = idx >> 6;
    const int f = idx & 63;
    out[idx] = bias[f];
    if (f == 0) { m[i] = -__builtin_inff(); z[i] = 0.0f; }
}

// ---------------- edge pass A: segment max of leaky_relu(es[src]+ed[dst]) ----------------
__global__ __launch_bounds__(256) void edge_max(const float* __restrict__ es,
                                                const float* __restrict__ ed,
                                                const int* __restrict__ src,
                                                const int* __restrict__ dst,
                                                float* __restrict__ m, int E, int n) {
    const int e = blockIdx.x * blockDim.x + threadIdx.x;
    if (e >= E + n) return;
    if (e + 16384 < E) {                      // stream the index arrays ahead of use
        __builtin_prefetch(src + e + 16384, 0, 1);
        __builtin_prefetch(dst + e + 16384, 0, 1);
    }
    int s, d;
    if (e < E) { s = src[e]; d = dst[e]; } else { s = d = e - E; }
    atomicMaxFloat(m + d, lrelu(es[s] + ed[d]));
}

// ---------------- edge pass B: z[dst] += exp(e - m[dst]) ----------------
__global__ __launch_bounds__(256) void edge_sum(const float* __restrict__ es,
                                                const float* __restrict__ ed,
                                                const int* __restrict__ src,
                                                const int* __restrict__ dst,
                                                const float* __restrict__ m,
                                                float* __restrict__ z, int E, int n) {
    const int e = blockIdx.x * blockDim.x + threadIdx.x;
    if (e >= E + n) return;
    int s, d;
    if (e < E) { s = src[e]; d = dst[e]; } else { s = d = e - E; }
    const float v = lrelu(es[s] + ed[d]);
    atomicAdd(z + d, __expf(v - m[d]));
}

// ---------------- edge pass C: out[dst] += alpha * hp[src]  (one wave per edge) ----------------
__global__ __launch_bounds__(256) void edge_aggr(const float* __restrict__ es,
                                                 const float* __restrict__ ed,
                                                 const int* __restrict__ src,
                                                 const int* __restrict__ dst,
                                                 const float* __restrict__ m,
                                                 const float* __restrict__ z,
                                                 const float* __restrict__ hp,
                                                 float* __restrict__ out, int E, int n) {
    const int e = (blockIdx.x * blockDim.x + threadIdx.x) >> 5;
    const int lane = threadIdx.x & 31;
    if (e >= E + n) return;
    int s, d;
    if (e < E) { s = src[e]; d = dst[e]; } else { s = d = e - E; }
    const float v = lrelu(es[s] + ed[d]);
    const float alpha = __expf(v - m[d]) / z[d];
    const float* hr = hp + (size_t)s * HID;
    float* o = out + (size_t)d * HID;
    atomicAdd(o + lane,      hr[lane]      * alpha);
    atomicAdd(o + lane + 32, hr[lane + 32] * alpha);
}

// ---------------- BN stats: per-channel sum & sumsq via LDS atomics ----------------
__global__ __launch_bounds__(256) void bn_stats(const float* __restrict__ x,
                                                float* __restrict__ stats, int n) {
    __shared__ float ssum[64];
    __shared__ float ssq[64];
    const int t = threadIdx.x;
    if (t < 64) { ssum[t] = 0.0f; ssq[t] = 0.0f; }
    __syncthreads();
    const int total = n * HID;
    for (int idx = blockIdx.x * blockDim.x + t; idx < total; idx += gridDim.x * blockDim.x) {
        const float v = x[idx];
        const int c = idx & 63;
        atomicAdd(&ssum[c], v);
        atomicAdd(&ssq[c], v * v);
    }
    __syncthreads();
    if (t < 64) {
        atomicAdd(&stats[t], ssum[t]);
        atomicAdd(&stats[64 + t], ssq[t]);
    }
}

// ---------------- BN apply + ReLU ----------------
__global__ __launch_bounds__(256) void bn_apply(const float* __restrict__ x,
                                                const float* __restrict__ stats,
                                                const float* __restrict__ g,
                                                const float* __restrict__ bta,
                                                float* __restrict__ y, int n) {
    const int idx = blockIdx.x * blockDim.x + threadIdx.x;
    if (idx >= n * HID) return;
    const int c = idx & 63;
    const float invn = 1.0f / (float)n;
    const float mu = stats[c] * invn;
    const float var = stats[64 + c] * invn - mu * mu;
    const float r = rsqrtf(var + BN_EPS);
    const float v = (x[idx] - mu) * r * g[c] + bta[c];
    y[idx] = v > 0.0f ? v : 0.0f;
}

// ---------------- pooling ----------------
__global__ void pool_init(float* __restrict__ pooled, float* __restrict__ cnt) {
    const int idx = blockIdx.x * blockDim.x + threadIdx.x;
    if (idx < N_GRAPHS * HID) pooled[idx] = 0.0f;
    if (idx < N_GRAPHS) cnt[idx] = 0.0f;
}

__global__ __launch_bounds__(256) void pool_kernel(const float* __restrict__ h,
                                                   const int* __restrict__ batch,
                                                   float* __restrict__ pooled,
                                                   float* __restrict__ cnt, int n) {
    const int warp = (blockIdx.x * blockDim.x + threadIdx.x) >> 5;
    const int lane = threadIdx.x & 31;
    if (warp >= n) return;
    const int g = batch[warp];
    const float* hr = h + (size_t)warp * HID;
    atomicAdd(&pooled[g * HID + lane],      hr[lane]);
    atomicAdd(&pooled[g * HID + lane + 32], hr[lane + 32]);
    if (lane == 0) atomicAdd(&cnt[g], 1.0f);
}

// ---------------- final MLP: 64 -> 50 -> 25 -> 1 (one thread per graph) ----------------
__global__ void mlp_kernel(const float* __restrict__ pooled, const float* __restrict__ cnt,
                           const float* __restrict__ W1, const float* __restrict__ b1,
                           const float* __restrict__ W2, const float* __restrict__ b2,
                           const float* __restrict__ W3, const float* __restrict__ b3,
                           float* __restrict__ out) {
    const int g = threadIdx.x;
    if (g >= N_GRAPHS) return;
    float hin[64];
    const float c = fmaxf(cnt[g], 1.0f);
    #pragma unroll
    for (int f = 0; f < 64; ++f) hin[f] = pooled[g * HID + f] / c;
    float h1[50];
    for (int j = 0; j < 50; ++j) {
        float s = b1[j];
        for (int f = 0; f < 64; ++f) s += hin[f] * W1[f * 50 + j];
        h1[j] = s > 0.0f ? s : 0.0f;
    }
    float h2[25];
    for (int k = 0; k < 25; ++k) {
        float s = b2[k];
        for (int j = 0; j < 50; ++j) s += h1[j] * W2[j * 25 + k];
        h2[k] = s > 0.0f ? s : 0.0f;
    }
    float o = b3[0];
    for (int k = 0; k < 25; ++k) o += h2[k] * W3[k];
    out[g] = o;
}

// ---------------- launcher ----------------
extern "C" void kernel_launch(void* const* d_in, const int* in_sizes, int n_in,
                              void* d_out, int out_size, void* d_ws, size_t ws_size,
                              hipStream_t stream) {
    const float* x     = (const float*)d_in[0];
    const float* W0    = (const float*)d_in[1];
    const float* Ws    = (const float*)d_in[2];
    const float* att_s = (const float*)d_in[3];
    const float* att_d = (const float*)d_in[4];
    const float* bias  = (const float*)d_in[5];
    const float* gamma = (const float*)d_in[6];
    const float* beta  = (const float*)d_in[7];
    const float* W1    = (const float*)d_in[8];
    const float* b1    = (const float*)d_in[9];
    const float* W2    = (const float*)d_in[10];
    const float* b2    = (const float*)d_in[11];
    const float* W3    = (const float*)d_in[12];
    const float* b3    = (const float*)d_in[13];
    const int*   eidx  = (const int*)d_in[14];
    const int*   batch = (const int*)d_in[15];
    float* out = (float*)d_out;

    const int n = N_NODES, E = N_EDGES;
    const int* src = eidx;
    const int* dst = eidx + E;

    // workspace carve-up (floats)
    float* ws = (float*)d_ws;
    size_t off = 0;
    float* hp    = ws + off; off += (size_t)n * HID;
    float* aggr  = ws + off; off += (size_t)n * HID;
    float* hbuf  = ws + off; off += (size_t)n * HID;
    float* es    = ws + off; off += n;
    float* ed    = ws + off; off += n;
    float* mseg  = ws + off; off += n;
    float* zseg  = ws + off; off += n;
    float* stats = ws + off; off += 128;
    float* pooled= ws + off; off += N_GRAPHS * HID;
    float* cnt   = ws + off; off += N_GRAPHS;
    (void)ws_size; (void)n_in; (void)in_sizes; (void)out_size;

    const int nTiles   = (n + 15) / 16;                  // 6250 waves
    const int gemmBlk  = (nTiles + 7) / 8;               // 8 waves per 256-thread block
    const int dotsBlk  = ((size_t)n * 32 + 255) / 256;
    const int elemBlk  = ((size_t)n * HID + 255) / 256;
    const int edgeBlk  = (E + n + 255) / 256;
    const int aggrBlk  = ((size_t)(E + n) * 32 + 255) / 256;
    const int poolBlk  = ((size_t)n * 32 + 255) / 256;

    const float* cur = x;
    for (int l = 0; l < N_LAYERS; ++l) {
        const float* W   = (l == 0) ? W0 : (Ws + (size_t)(l - 1) * HID * HID);
        const float* a_s = att_s + (size_t)l * HID;
        const float* a_d = att_d + (size_t)l * HID;
        const float* bs  = bias  + (size_t)l * HID;
        const float* gm  = gamma + (size_t)l * HID;
        const float* bt  = beta  + (size_t)l * HID;

        gemm64_wmma<<<gemmBlk, 256, 0, stream>>>(cur, W, hp, n);
        dots_kernel<<<dotsBlk, 256, 0, stream>>>(hp, a_s, a_d, es, ed, n);
        init_layer<<<elemBlk, 256, 0, stream>>>(mseg, zseg, aggr, bs, stats, n);
        edge_max<<<edgeBlk, 256, 0, stream>>>(es, ed, src, dst, mseg, E, n);
        edge_sum<<<edgeBlk, 256, 0, stream>>>(es, ed, src, dst, mseg, zseg, E, n);
        edge_aggr<<<aggrBlk, 256, 0, stream>>>(es, ed, src, dst, mseg, zseg, hp, aggr, E, n);
        bn_stats<<<512, 256, 0, stream>>>(aggr, stats, n);
        bn_apply<<<elemBlk, 256, 0, stream>>>(aggr, stats, gm, bt, hbuf, n);
        cur = hbuf;
    }

    pool_init<<<(N_GRAPHS * HID + 255) / 256, 256, 0, stream>>>(pooled, cnt);
    pool_kernel<<<poolBlk, 256, 0, stream>>>(cur, batch, pooled, cnt, n);
    mlp_kernel<<<1, 64, 0, stream>>>(pooled, cnt, W1, b1, W2, b2, W3, b3, out);
}